// ImageWarp_3822520893488
// MI455X (gfx1250) — compile-verified
//
#include <hip/hip_runtime.h>

// ImageWarp (crop+resize gather): out[b,c,i,j] = im[rows[b,j], cols[b,i], c]
// im: (1024,1024,3) f32, rects: (1024,4) i32, out: (1024,3,227,227) f32.
// Store-bandwidth bound (~633 MB out); image (12 MB) lives in L2 (192 MB).

#define CNN 227
#define DIL 16
#define NPOS (CNN * CNN)   // 51529 positions per (b, plane)
#define IMW 1024
#define IMH 1024

__global__ __launch_bounds__(256) void imagewarp_gather_kernel(
    const float* __restrict__ im,
    const int*   __restrict__ rects,
    float*       __restrict__ out,
    int nrect)
{
    __shared__ int rowOff[CNN];  // row * (1024*3)  : float offset of row start
    __shared__ int colOff[CNN];  // col * 3         : float offset within row

    const int b   = blockIdx.x;
    const int tid = threadIdx.x;
    if (b >= nrect) return;

    // ---- rect -> sampling grid (matches reference integer math) ----
    const int x = rects[b * 4 + 0];
    const int y = rects[b * 4 + 1];
    const int w = rects[b * 4 + 2];
    const int h = rects[b * 4 + 3];

    const int x_min = max(0,   x - DIL);
    const int x_max = min(IMW, x + w + DIL);
    const int y_min = max(0,   y - DIL);
    const int y_max = min(IMH, y + h + DIL);
    const int dx = x_max - x_min;   // >= 1 (non-negative by construction)
    const int dy = y_max - y_min;

    if (tid < CNN) {
        // cols[i] = clip(x_min + i*dx/227, 0, W-1); floor-div, operands >= 0
        int c = x_min + (tid * dx) / CNN;
        c = min(max(c, 0), IMW - 1);
        colOff[tid] = c * 3;

        int r = y_min + (tid * dy) / CNN;
        r = min(max(r, 0), IMH - 1);
        rowOff[tid] = r * (IMW * 3);
    }
    __syncthreads();

    // ---- gather + streaming stores ----
    float* __restrict__ outb = out + (size_t)b * (3 * NPOS);

    // split the 51529-position space across blockIdx.y chunks
    const int nchunk = gridDim.y;
    const int chunk  = (NPOS + nchunk - 1) / nchunk;
    const int p0     = blockIdx.y * chunk;
    const int p1     = min(NPOS, p0 + chunk);

    for (int p = p0 + tid; p < p1; p += 256) {
        const int i = p / CNN;          // constant divisor -> mulhi, no div
        const int j = p - i * CNN;

        // 12-byte gather of one pixel (r,g,b) -> global_load_b96; stays in L2
        const float* __restrict__ src = im + (rowOff[j] + colOff[i]);
        const float vr = src[0];
        const float vg = src[1];
        const float vb = src[2];

        // p == i*227 + j is exactly the per-plane output offset.
        // NT stores: don't let the 633 MB output stream evict the image from L2.
        __builtin_nontemporal_store(vr, outb + p);
        __builtin_nontemporal_store(vg, outb + NPOS + p);
        __builtin_nontemporal_store(vb, outb + 2 * NPOS + p);
    }
}

extern "C" void kernel_launch(void* const* d_in, const int* in_sizes, int n_in,
                              void* d_out, int out_size, void* d_ws, size_t ws_size,
                              hipStream_t stream) {
    const float* im    = (const float*)d_in[0];   // (1024,1024,3) f32
    const int*   rects = (const int*)  d_in[1];   // (1024,4) i32
    float*       out   = (float*)d_out;           // (1024,3,227,227) f32

    const int nrect = in_sizes[1] / 4;            // 1024

    // 4 chunks per rect: 4096 blocks x 8 wave32 -> plenty of latency hiding
    dim3 grid(nrect, 4, 1);
    imagewarp_gather_kernel<<<grid, 256, 0, stream>>>(im, rects, out, nrect);
}